// gumbelq_69114613727242
// MI455X (gfx1250) — compile-verified
//
#include <hip/hip_runtime.h>

typedef float v2f __attribute__((ext_vector_type(2)));
typedef float v8f __attribute__((ext_vector_type(8)));

#define BS_TOTAL 2048      // B*S
#define GV       640       // NUM_GROUPS*NUM_VARS
#define HDIM     512       // H
#define NVARS    320
#define NGRP     2
#define CDIM     128       // CODE_DIM / NUM_GROUPS
#define MTILES   128       // BS_TOTAL/16
#define NTILES   40        // GV/16

// ---------------------------------------------------------------- zero marg
__global__ __launch_bounds__(256) void gq_zero_marg(float* __restrict__ marg) {
    int i = blockIdx.x * 256 + threadIdx.x;
    if (i < NGRP * NVARS) marg[i] = 0.0f;
}

// ---------------------------------------------------------------- WMMA GEMM
// logits[m][n] = sum_k x[m][k] * W[k][n] + b[n]
// One wave per 16x16 tile, K stepped by 4 via V_WMMA_F32_16X16X4_F32.
__global__ __launch_bounds__(256) void gq_gemm_wmma(const float* __restrict__ X,
                                                    const float* __restrict__ Wm,
                                                    const float* __restrict__ bias,
                                                    float* __restrict__ logits) {
    const int wid  = (blockIdx.x * 256 + threadIdx.x) >> 5;   // global wave id
    const int lane = threadIdx.x & 31;
    const int nt   = wid % NTILES;
    const int mt   = wid / NTILES;
    const int hi   = lane >> 4;        // half-wave selector
    const int lo   = lane & 15;

    const int arow = mt * 16 + lo;     // A: lanes 0-15 -> M, K=0/1 ; 16-31 -> M, K=2/3
    const int bcol = nt * 16 + lo;     // B: lanes 0-15 -> N, K=0/1 ; 16-31 -> N, K=2/3

    const float* aptr = X  + (size_t)arow * HDIM + 2 * hi;
    const float* bptr = Wm + (size_t)(2 * hi) * GV + bcol;

    v8f acc = {};
#pragma unroll 4
    for (int k = 0; k < HDIM; k += 4) {
        v2f a = *(const v2f*)(aptr + k);                 // global_load_b64
        v2f b;
        b.x = bptr[(size_t)k * GV];
        b.y = bptr[(size_t)k * GV + GV];
        acc = __builtin_amdgcn_wmma_f32_16x16x4_f32(
            /*neg_a=*/false, a, /*neg_b=*/false, b,
            /*c_mod=*/(short)0, acc, /*reuse_a=*/false, /*reuse_b=*/false);
    }

    // C/D layout: VGPR r, lanes 0-15 -> M = r, lanes 16-31 -> M = r + 8
    const float bv    = bias[bcol];
    const int   rbase = mt * 16 + hi * 8;
#pragma unroll
    for (int r = 0; r < 8; ++r)
        logits[(size_t)(rbase + r) * GV + bcol] = acc[r] + bv;
}

// ---------------------------------------------------------------- per-row op
// One wave per row of l [4096 x 320]: gumbel-softmax-hard + codevector gather,
// plus atomic accumulation of the raw-softmax marginal.
__global__ __launch_bounds__(256) void gq_rows(const float* __restrict__ logits,
                                               const float* __restrict__ gumbels,
                                               const float* __restrict__ codev,
                                               float* __restrict__ out,
                                               float* __restrict__ marg) {
    const int wid  = (blockIdx.x * 256 + threadIdx.x) >> 5;   // row id [0,4096)
    const int lane = threadIdx.x & 31;
    const int bs = wid >> 1;
    const int g  = wid & 1;

    const float* lrow = logits  + (size_t)bs * GV + g * NVARS;
    const float* grow = gumbels + (size_t)wid * NVARS;

    float lv[10], zv[10];
    float zmax = -__builtin_inff();
    int   zidx = 0;
    float lmax = -__builtin_inff();
#pragma unroll
    for (int j = 0; j < 10; ++j) {
        const int v = lane + 32 * j;
        const float l = lrow[v];
        const float z = (l + grow[v]) * 0.5f;             // /TAU, TAU=2
        lv[j] = l; zv[j] = z;
        if (z > zmax) { zmax = z; zidx = v; }
        lmax = fmaxf(lmax, l);
    }
    // wave32 reductions: argmax (first-index tie-break) and max
#pragma unroll
    for (int off = 16; off > 0; off >>= 1) {
        float om = __shfl_xor(zmax, off, 32);
        int   oi = __shfl_xor(zidx, off, 32);
        if (om > zmax || (om == zmax && oi < zidx)) { zmax = om; zidx = oi; }
        lmax = fmaxf(lmax, __shfl_xor(lmax, off, 32));
    }

    float se = 0.0f, sl = 0.0f;
#pragma unroll
    for (int j = 0; j < 10; ++j) {
        se += expf(zv[j] - zmax);
        sl += expf(lv[j] - lmax);
    }
#pragma unroll
    for (int off = 16; off > 0; off >>= 1) {
        se += __shfl_xor(se, off, 32);
        sl += __shfl_xor(sl, off, 32);
    }

    // straight-through value: (1 - p_argmax) + p_argmax, p_argmax = 1/se
    const float pmax = 1.0f / se;
    const float cp   = (1.0f - pmax) + pmax;

    // marginal of raw softmax (no gumbel, no tau)
    const float inv_sl = 1.0f / sl;
#pragma unroll
    for (int j = 0; j < 10; ++j)
        atomicAdd(&marg[g * NVARS + lane + 32 * j], expf(lv[j] - lmax) * inv_sl);

    // gather selected codevector row [128], scale by cp, write output
    const float4* cvr = (const float4*)(codev + (size_t)(g * NVARS + zidx) * CDIM);
    float4 c = cvr[lane];
    c.x *= cp; c.y *= cp; c.z *= cp; c.w *= cp;
    ((float4*)(out + (size_t)bs * (NGRP * CDIM) + g * CDIM))[lane] = c;
}

// ---------------------------------------------------------------- perplexity
__global__ __launch_bounds__(32) void gq_finalize(const float* __restrict__ marg,
                                                  float* __restrict__ perp_out) {
    const int lane = threadIdx.x;
    float perp = 0.0f;
#pragma unroll
    for (int g = 0; g < NGRP; ++g) {
        float acc = 0.0f;
#pragma unroll
        for (int j = 0; j < 10; ++j) {
            const float m = marg[g * NVARS + lane + 32 * j] * (1.0f / (float)BS_TOTAL);
            acc += m * logf(m + 1e-7f);
        }
#pragma unroll
        for (int off = 16; off > 0; off >>= 1)
            acc += __shfl_xor(acc, off, 32);
        perp += expf(-acc);
    }
    if (lane == 0) perp_out[0] = perp;
}

// ---------------------------------------------------------------- launcher
extern "C" void kernel_launch(void* const* d_in, const int* in_sizes, int n_in,
                              void* d_out, int out_size, void* d_ws, size_t ws_size,
                              hipStream_t stream) {
    const float* x       = (const float*)d_in[0];   // [4,512,512]
    const float* Wm      = (const float*)d_in[1];   // [512,640]
    const float* bias    = (const float*)d_in[2];   // [640]
    const float* codev   = (const float*)d_in[3];   // [1,640,128]
    const float* gumbels = (const float*)d_in[4];   // [4096,320]
    float* out = (float*)d_out;                     // [4,512,256] + perp scalar

    float* logits = (float*)d_ws;                   // 2048*640 f32 = 5.24 MB
    float* marg   = logits + (size_t)BS_TOTAL * GV; // 640 f32

    gq_zero_marg<<<(NGRP * NVARS + 255) / 256, 256, 0, stream>>>(marg);
    gq_gemm_wmma<<<(MTILES * NTILES) / 8, 256, 0, stream>>>(x, Wm, bias, logits);
    gq_rows<<<(BS_TOTAL * NGRP) / 8, 256, 0, stream>>>(logits, gumbels, codev, out, marg);
    gq_finalize<<<1, 32, 0, stream>>>(marg, out + (size_t)BS_TOTAL * NGRP * CDIM);
}